// CorpBEVT_46892452937730
// MI455X (gfx1250) — compile-verified
//
#include <hip/hip_runtime.h>
#include <stdint.h>

// Problem constants (match reference): B=2, L=5, C=128, H=256, W=256, S=32768
#define Bn 2
#define Ln 5
#define Cn 128
#define Hn 256
#define Wn 256
#define Sn 32768
#define HWn   (Hn * Wn)          // 65536 = 2^16
#define PLANES (Bn * Ln)         // 10
#define PLANE_ELEMS ((unsigned)Cn * HWn)         // 8,388,608 = 2^23
#define TOTAL_ELEMS ((unsigned long long)PLANES * PLANE_ELEMS)  // 83,886,080
#define TOTAL_VEC   (TOTAL_ELEMS / 4)            // 20,971,520 float4s
#define SEL_BLOCK   256          // TOTAL_VEC % SEL_BLOCK == 0 (81,920 blocks)

typedef float f4 __attribute__((ext_vector_type(4)));

// ---------------------------------------------------------------------------
// Kernel 1: zero the flag planes (d_ws is poisoned by the harness, re-zero
// every call; 10 * 65536 bytes = 655,360 B = 163,840 dwords).
// ---------------------------------------------------------------------------
__global__ void CorpBEVT_zero_flags(uint32_t* __restrict__ flags32, int n) {
    int i = blockIdx.x * blockDim.x + threadIdx.x;
    if (i < n) flags32[i] = 0u;
}

// ---------------------------------------------------------------------------
// Kernel 2: invert the scatter. One thread per (b*L+l, i) pair. Truncation
// via C cast matches jnp .astype(int32) (round toward zero). Duplicate
// positions all store the same byte value -> benign race.
// ---------------------------------------------------------------------------
__global__ void CorpBEVT_mark_flags(const float* __restrict__ grid,   // [PLANES, HW, 2]
                                    const int*   __restrict__ sel,    // [S]
                                    uint8_t*     __restrict__ flags)  // [PLANES, HW]
{
    int t = blockIdx.x * blockDim.x + threadIdx.x;
    if (t >= PLANES * Sn) return;
    int bl = t >> 15;             // / Sn  (Sn = 2^15)
    int i  = t & (Sn - 1);        // % Sn
    int s  = sel[i];
    const float* g = grid + (((size_t)bl * HWn + (size_t)s) << 1);
    float gx = g[0];
    float gy = g[1];
    int xi = (int)gx;             // trunc toward zero
    int yi = (int)gy;
    if (xi >= 0 && xi < Hn && yi >= 0 && yi < Wn) {
        flags[((size_t)bl << 16) + ((size_t)xi << 8) + (size_t)yi] = 1;
    }
}

// ---------------------------------------------------------------------------
// Kernel 3: single fully-coalesced streaming pass over all 83.9M floats.
// Per 256-thread block:
//   - async-stage the block's contiguous 1KB flag slice into LDS via
//     global_load_async_to_lds_b32 (CDNA5 async engine, ASYNCcnt),
//   - overlap with 2x NT global_load_b128 of the big streams,
//   - s_wait_asynccnt 0 + barrier, ds_load the flags from LDS, select per
//     component, NT global_store_b128.
// All index math is shifts/ands (powers of two). A block's 1024 elements
// never cross a plane boundary (2^23 % 1024 == 0), so its flag slice is
// contiguous. Launch covers TOTAL_VEC exactly (no guard, no barrier
// divergence, all lanes active for the per-lane async issue).
// ---------------------------------------------------------------------------
__global__ void CorpBEVT_select_copy(const f4* __restrict__ outp,
                                     const f4* __restrict__ basep,
                                     const uint8_t* __restrict__ flags,
                                     f4* __restrict__ dst)
{
    __shared__ uint32_t sflags[SEL_BLOCK];     // 1KB flag slice for this block

    unsigned v  = blockIdx.x * SEL_BLOCK + threadIdx.x;   // float4 index
    unsigned e  = v << 2;                 // element index (< 2^27)
    unsigned bl = e >> 23;                // / (C*HW)
    unsigned p  = e & (HWn - 1);          // spatial position of lane's 1st elem

    // --- CDNA5 async global->LDS staging of 4 flag bytes per lane ---------
    // LDS dest address is taken from the VDST VGPR; low 32 bits of a flat LDS pointer are
    // the 0-based offset within the workgroup's LDS allocation.
    uint32_t lds_addr = (uint32_t)(uintptr_t)(&sflags[threadIdx.x]);
    uint32_t gvoff    = (bl << 16) + p;   // byte offset from flags base (p%4==0)
    uint64_t gbase    = (uint64_t)(uintptr_t)flags;
    asm volatile("global_load_async_to_lds_b32 %0, %1, %2"
                 :
                 : "v"(lds_addr), "v"(gvoff), "s"(gbase)
                 : "memory");

    // Overlap the two big NT streams with the async flag copy.
    f4 b = __builtin_nontemporal_load(&basep[v]);
    f4 o = __builtin_nontemporal_load(&outp[v]);

    asm volatile("s_wait_asynccnt 0x0" ::: "memory");
    __syncthreads();

    // Plain addrspace(3) read -> ds_load_b32. The asm "memory" clobber +
    // barrier prevent the compiler from assuming LDS is unwritten.
    uint32_t f = sflags[threadIdx.x];

    f4 r;
    r.x = (f & 0x000000FFu) ? o.x : b.x;  // little-endian: byte0 = pos p
    r.y = (f & 0x0000FF00u) ? o.y : b.y;
    r.z = (f & 0x00FF0000u) ? o.z : b.z;
    r.w = (f & 0xFF000000u) ? o.w : b.w;
    __builtin_nontemporal_store(r, &dst[v]);
}

// ---------------------------------------------------------------------------
// Fallback path (only if d_ws is too small to hold the 640KB flag planes):
// plain copy + direct strided scatter.
// ---------------------------------------------------------------------------
__global__ void CorpBEVT_copy_base(const f4* __restrict__ basep,
                                   f4* __restrict__ dst)
{
    unsigned v = blockIdx.x * blockDim.x + threadIdx.x;
    if (v >= (unsigned)TOTAL_VEC) return;
    __builtin_nontemporal_store(__builtin_nontemporal_load(&basep[v]), &dst[v]);
}

__global__ void CorpBEVT_scatter_direct(const float* __restrict__ outp,
                                        const float* __restrict__ grid,
                                        const int*   __restrict__ sel,
                                        float*       __restrict__ dst)
{
    int t = blockIdx.x * blockDim.x + threadIdx.x;   // one thread per (bl, i)
    if (t >= PLANES * Sn) return;
    int bl = t >> 15;
    int i  = t & (Sn - 1);
    int s  = sel[i];
    const float* g = grid + (((size_t)bl * HWn + (size_t)s) << 1);
    int xi = (int)g[0];
    int yi = (int)g[1];
    if (xi < 0 || xi >= Hn || yi < 0 || yi >= Wn) return;
    size_t flat = ((size_t)xi << 8) + (size_t)yi;
    size_t base_off = (size_t)bl * PLANE_ELEMS + flat;
    // duplicate (bl,i) with same flat write identical values -> benign
    #pragma unroll 4
    for (int c = 0; c < Cn; ++c) {
        size_t off = base_off + ((size_t)c << 16);
        dst[off] = outp[off];
    }
}

// ---------------------------------------------------------------------------
extern "C" void kernel_launch(void* const* d_in, const int* in_sizes, int n_in,
                              void* d_out, int out_size, void* d_ws, size_t ws_size,
                              hipStream_t stream)
{
    const float* outp  = (const float*)d_in[0];   // output  [B,L,C,H,W]
    const float* basep = (const float*)d_in[1];   // base    [B,L,C,H,W]
    const float* grid  = (const float*)d_in[2];   // grid    [B,L,H,W,2]
    const int*   sel   = (const int*)d_in[3];     // indices [S]
    float*       dst   = (float*)d_out;

    const size_t flag_bytes = (size_t)PLANES * HWn;    // 655,360 B

    if (ws_size >= flag_bytes) {
        uint8_t* flags = (uint8_t*)d_ws;

        const int nflag32 = (int)(flag_bytes / 4);     // 163,840
        CorpBEVT_zero_flags<<<(nflag32 + 255) / 256, 256, 0, stream>>>(
            (uint32_t*)flags, nflag32);

        const int nmark = PLANES * Sn;                 // 327,680
        CorpBEVT_mark_flags<<<(nmark + 255) / 256, 256, 0, stream>>>(
            grid, sel, flags);

        const unsigned nblocks = (unsigned)(TOTAL_VEC / SEL_BLOCK);  // 81,920
        CorpBEVT_select_copy<<<nblocks, SEL_BLOCK, 0, stream>>>(
            (const f4*)outp, (const f4*)basep, flags, (f4*)dst);
    } else {
        const unsigned nvec = (unsigned)TOTAL_VEC;
        CorpBEVT_copy_base<<<(nvec + 255) / 256, 256, 0, stream>>>(
            (const f4*)basep, (f4*)dst);

        const int nmark = PLANES * Sn;
        CorpBEVT_scatter_direct<<<(nmark + 255) / 256, 256, 0, stream>>>(
            outp, grid, sel, dst);
    }
}